// KAN_attention_16655883173904
// MI455X (gfx1250) — compile-verified
//
#include <hip/hip_runtime.h>
#include <hip/hip_bf16.h>
#include <stdint.h>

typedef __attribute__((ext_vector_type(2))) float v2f;
typedef __attribute__((ext_vector_type(4))) float v4f;
typedef __attribute__((ext_vector_type(8))) float v8f;
typedef __attribute__((ext_vector_type(4))) unsigned int u32x4;
typedef __attribute__((ext_vector_type(4))) int i32x4;
typedef __attribute__((ext_vector_type(8))) int i32x8;

#define ACT 2048      // act_in == act_out
#define BATCH 8
#define CCH 8         // spline coefficients per edge
#define NCH 12        // padded channels per input: 8 spline + silu + 3 zeros (3 k-steps of 4)
#define OT 16         // output columns per WMMA tile
#define WAVES 2       // waves per block
#define IW 32         // inputs (i) per wave
#define IB (WAVES * IW)   // inputs per block = 64
#define ICH (ACT / IB)    // 32 i-chunks
#define NSLOT (ICH * 2)   // 64 partial slots (i-chunk x branch)
#define CO_STRIDE 260     // 256 data dwords + 4 pad dwords per o-row in LDS
#define A_I_STRIDE (3 * 64)  // A_ws floats per (branch,i): 3 k-steps x 32 lanes x 2

// ---------------------------------------------------------------------------
// Kernel 1: B-spline basis + silu, stored in WMMA A-operand lane layout.
// A 16x4 f32 layout: lanes 0-15 -> M=lane, VGPR0/1 = K0/K1; lanes 16-31 -> K2/K3.
// ---------------------------------------------------------------------------
__global__ void kan_basis_kernel(const float* __restrict__ q,
                                 const float* __restrict__ k,
                                 float* __restrict__ A_ws) {
  int tid = blockIdx.x * blockDim.x + threadIdx.x;   // 2*2048*16 = 65536
  int branch = tid >> 15;
  int rem = tid & 32767;
  int i = rem >> 4;
  int row = rem & 15;          // M row (batch index; rows 8..15 are zero padding)

  float a[NCH];
#pragma unroll
  for (int c = 0; c < NCH; ++c) a[c] = 0.0f;

  if (row < BATCH) {
    const float* xsrc = branch ? k : q;
    float x = xsrc[row * ACT + i];
    // Extended knots: t_j = -2.2 + 0.4*j, j = 0..11 (pykan grid extend, POLY=3)
    float B[11];
#pragma unroll
    for (int j = 0; j < 11; ++j) {
      float t0 = -2.2f + 0.4f * (float)j;
      B[j] = (x >= t0 && x < t0 + 0.4f) ? 1.0f : 0.0f;
    }
#pragma unroll
    for (int d = 1; d <= 3; ++d) {
      float inv = 1.0f / (0.4f * (float)d);
#pragma unroll 10
      for (int j = 0; j + d < 11; ++j) {
        float tj = -2.2f + 0.4f * (float)j;
        float left  = (x - tj) * inv;
        float right = ((tj + 0.4f * (float)(d + 1)) - x) * inv;
        B[j] = left * B[j] + right * B[j + 1];
      }
    }
#pragma unroll
    for (int c = 0; c < 8; ++c) a[c] = B[c];
    a[8] = x / (1.0f + expf(-x));   // silu base activation channel
  }

  size_t base = (size_t)(branch * ACT + i) * A_I_STRIDE;
#pragma unroll
  for (int c = 0; c < NCH; ++c) {
    int t = c >> 2, kk = c & 3;
    int lane = row + ((kk >= 2) ? 16 : 0);
    int v = kk & 1;
    A_ws[base + (size_t)t * 64 + lane * 2 + v] = a[c];
  }
}

// ---------------------------------------------------------------------------
// Kernel 2: streaming WMMA GEMM over the 268 MB coefficient tensor.
// Each wave TDM-loads its 16 x (32i x 8c) coef subtile (row stride 64 KB) into
// LDS with +4-dword row padding (bank-conflict-free ds_load_b64 gathers), then
// runs 3 independent v_wmma_f32_16x16x4_f32 accumulator chains per input i.
// ---------------------------------------------------------------------------
__global__ void __launch_bounds__(WAVES * 32)
kan_gemm_kernel(const float* __restrict__ coef_q, const float* __restrict__ coef_k,
                const float* __restrict__ scale_base, const float* __restrict__ scale_sp,
                const float* __restrict__ mask_q, const float* __restrict__ mask_k,
                const float* __restrict__ A_ws, float* __restrict__ y_part) {
  __shared__ float coef_lds[WAVES][OT][CO_STRIDE];   // 33.3 KB
  __shared__ float sb_lds[WAVES][OT][IW + 1];        //  4.2 KB
  __shared__ float sp_lds[WAVES][OT][IW + 1];        //  4.2 KB
  __shared__ float part_lds[WAVES][256];             //  2.0 KB

  const int tid = threadIdx.x;
  const int lane = tid & 31;
  const int wave = __builtin_amdgcn_readfirstlane(tid) >> 5;   // scalar wave id
  const int o_base = blockIdx.x * OT;
  const int branch = blockIdx.z;
  const int i_wv = blockIdx.y * IB + wave * IW;

  const float* coef = branch ? coef_k : coef_q;
  const float* mask = branch ? mask_k : mask_q;

  // --- stage sb = mask*scale_base, sp = mask*scale_sp (coalesced in i) ---
  for (int o = 0; o < OT; ++o) {
    size_t g = (size_t)(o_base + o) * ACT + (i_wv + lane);
    float m = mask[g];
    sb_lds[wave][o][lane] = m * scale_base[g];
    sp_lds[wave][o][lane] = m * scale_sp[g];
  }

  // --- TDM: one tensor_load_to_lds per wave for the coef subtile ---
  // Tile: 16 rows x 256 f32 elements, tensor row stride 16384 elements,
  // LDS pad: +4 dwords every 256 dwords -> LDS row stride 260 dwords.
  {
    unsigned lds_off = (unsigned)(uintptr_t)&coef_lds[wave][0][0];
    unsigned long long ga = (unsigned long long)(uintptr_t)coef +
        ((unsigned long long)((size_t)o_base * ACT + i_wv) * CCH) * 4ull;
    u32x4 g0;
    g0[0] = 1u;                                   // count = 1 (valid descriptor)
    g0[1] = lds_off;                              // LDS byte address
    g0[2] = (unsigned)ga;                         // global_addr[31:0]
    g0[3] = (unsigned)((ga >> 32) & 0x01FFFFFFu) | (2u << 30);  // addr[56:32] | type=2
    i32x8 g1;
    g1[0] = (int)((2u << 16)    // data_size = 4 B
                | (1u << 20)    // pad_enable
                | (7u << 22)    // pad_interval: 256 dwords
                | (3u << 25));  // pad_amount: 4 dwords
    g1[1] = (int)((16384u & 0xFFFFu) << 16);      // tensor_dim0 lo16 (=ACT*CCH)
    g1[2] = (int)((2048u & 0xFFFFu) << 16);       // tensor_dim0 hi | tensor_dim1 lo (=ACT rows)
    g1[3] = (int)(256u << 16);                    // tensor_dim1 hi | tile_dim0 = 256
    g1[4] = 16;                                   // tile_dim1 = 16 | tile_dim2 = 0
    g1[5] = 16384;                                // tensor_dim0_stride lo32
    g1[6] = 0;
    g1[7] = 0;
    i32x4 gz; gz[0] = gz[1] = gz[2] = gz[3] = 0;  // 2D tensor: groups 2/3 unused
#if __clang_major__ >= 23
    i32x8 gz8; gz8[0]=gz8[1]=gz8[2]=gz8[3]=gz8[4]=gz8[5]=gz8[6]=gz8[7]=0;
    __builtin_amdgcn_tensor_load_to_lds(g0, g1, gz, gz, gz8, 0);
#else
    __builtin_amdgcn_tensor_load_to_lds(g0, g1, gz, gz, 0);
#endif
    asm volatile("" ::: "memory");
    __builtin_amdgcn_s_wait_tensorcnt(0);
    asm volatile("" ::: "memory");
  }

  // --- WMMA main loop: D = A(16x4) x B(4x16) + C, 3 k-steps per input i ---
  // Three independent accumulator chains (one per k-step class) for WMMA ILP.
  v8f acc0 = {0.f, 0.f, 0.f, 0.f, 0.f, 0.f, 0.f, 0.f};
  v8f acc1 = acc0;
  v8f acc2 = acc0;
  const int n = lane & 15;                 // output column handled by this lane
  const float hisel = (lane >= 16) ? 0.0f : 1.0f;   // zero channels {10,11} side
  const int coff = (lane >= 16) ? 2 : 0;   // lanes 16-31 carry K+2/K+3
  const float* Abase = A_ws + (size_t)(branch * ACT + i_wv) * A_I_STRIDE + lane * 2;
  const float* crow = &coef_lds[wave][n][coff];

  for (int ii = 0; ii < IW; ++ii) {
    float sp  = sp_lds[wave][n][ii];
    float sbv = sb_lds[wave][n][ii] * hisel;     // LDS same-address broadcast for lanes>=16
    v2f c01 = *(const v2f*)(crow + ii * 8);      // channels {0,1} / {2,3}
    v2f c45 = *(const v2f*)(crow + ii * 8 + 4);  // channels {4,5} / {6,7}
    v2f b0; b0[0] = c01[0] * sp; b0[1] = c01[1] * sp;
    v2f b1; b1[0] = c45[0] * sp; b1[1] = c45[1] * sp;
    v2f b2; b2[0] = sbv; b2[1] = 0.0f;           // channels {8,9}/{10,11}: silu weight sb, zeros
    const float* ap = Abase + (size_t)ii * A_I_STRIDE;
    v2f a0 = *(const v2f*)(ap);
    v2f a1 = *(const v2f*)(ap + 64);
    v2f a2 = *(const v2f*)(ap + 128);
    acc0 = __builtin_amdgcn_wmma_f32_16x16x4_f32(false, a0, false, b0, (short)0, acc0, false, false);
    acc1 = __builtin_amdgcn_wmma_f32_16x16x4_f32(false, a1, false, b1, (short)0, acc1, false, false);
    acc2 = __builtin_amdgcn_wmma_f32_16x16x4_f32(false, a2, false, b2, (short)0, acc2, false, false);
  }

  // --- cross-wave reduction via LDS, then write partial tile ---
#pragma unroll
  for (int r = 0; r < 8; ++r)
    part_lds[wave][r * 32 + lane] = acc0[r] + acc1[r] + acc2[r];
  __syncthreads();
  if (wave == 0) {
    const int slot = blockIdx.y * 2 + branch;
    for (int idx = lane; idx < 256; idx += 32) {
      float s = part_lds[0][idx] + part_lds[1][idx];
      int r = idx >> 5, l = idx & 31;
      int M = r + ((l & 16) ? 8 : 0);     // C/D layout: VGPR r, lanes 16-31 -> M = r+8
      int N = l & 15;
      if (M < BATCH)
        y_part[((size_t)slot * BATCH + M) * ACT + o_base + N] = s;
    }
  }
}

// ---------------------------------------------------------------------------
// Kernel 3: reduce 64 partial slots + bias + softmax over out_dim = 8.
// ---------------------------------------------------------------------------
__global__ void kan_softmax_kernel(const float* __restrict__ y_part,
                                   const float* __restrict__ bias_w,
                                   float* __restrict__ out) {
  int t = blockIdx.x * blockDim.x + threadIdx.x;   // 8 * 256 = 2048
  int b = t >> 8;          // batch
  int gp = t & 255;        // (head, patch) group
  float v[8];
  const float* bp = bias_w + gp * 8;
#pragma unroll
  for (int j = 0; j < 8; ++j) v[j] = bp[j];
  for (int slot = 0; slot < NSLOT; ++slot) {
    const float* pp = y_part + ((size_t)slot * BATCH + b) * ACT + gp * 8;
    v4f p0 = *(const v4f*)pp;
    v4f p1 = *(const v4f*)(pp + 4);
    v[0] += p0[0]; v[1] += p0[1]; v[2] += p0[2]; v[3] += p0[3];
    v[4] += p1[0]; v[5] += p1[1]; v[6] += p1[2]; v[7] += p1[3];
  }
  float m = v[0];
#pragma unroll
  for (int j = 1; j < 8; ++j) m = fmaxf(m, v[j]);
  float s = 0.f;
#pragma unroll
  for (int j = 0; j < 8; ++j) { v[j] = expf(v[j] - m); s += v[j]; }
  float inv = 1.0f / s;
  float* op = out + (size_t)b * ACT + gp * 8;
#pragma unroll
  for (int j = 0; j < 8; ++j) op[j] = v[j] * inv;
}

extern "C" void kernel_launch(void* const* d_in, const int* in_sizes, int n_in,
                              void* d_out, int out_size, void* d_ws, size_t ws_size,
                              hipStream_t stream) {
  const float* q          = (const float*)d_in[0];
  const float* k          = (const float*)d_in[1];
  const float* coef_q     = (const float*)d_in[2];
  const float* coef_k     = (const float*)d_in[3];
  const float* scale_base = (const float*)d_in[4];
  const float* scale_sp   = (const float*)d_in[5];
  const float* mask_q     = (const float*)d_in[6];
  const float* mask_k     = (const float*)d_in[7];
  const float* bias_w     = (const float*)d_in[8];
  float* out = (float*)d_out;

  // workspace: A_ws = 2*2048*192 floats (3 MB), y_part = 64*8*2048 floats (4 MB)
  float* A_ws = (float*)d_ws;
  float* y_part = A_ws + (size_t)2 * ACT * A_I_STRIDE;

  kan_basis_kernel<<<256, 256, 0, stream>>>(q, k, A_ws);
  dim3 grid(ACT / OT, ICH, 2);
  kan_gemm_kernel<<<grid, WAVES * 32, 0, stream>>>(coef_q, coef_k, scale_base, scale_sp,
                                                   mask_q, mask_k, A_ws, y_part);
  kan_softmax_kernel<<<8, 256, 0, stream>>>(y_part, bias_w, out);
}